// TabularEmbedding_26972394619707
// MI455X (gfx1250) — compile-verified
//
#include <hip/hip_runtime.h>

// TabularEmbedding fused: out[b,i,d] = LN_d(x[b,i]*W[i,d] + b[i,d]) * gamma + beta
// Bandwidth-bound (1 GiB f32 output @ 23.3 TB/s -> ~46us floor).
// LayerNorm reduction removed via closed-form per-feature stats; each 16x16
// output tile is a K=2 GEMM done with V_WMMA_F32_16X16X4_F32 (K=2,3 zero-padded).

typedef __attribute__((ext_vector_type(2))) float v2f;
typedef __attribute__((ext_vector_type(8))) float v8f;

#define NF 64
#define ND 128
#define LN_EPS 1e-5f

// ---------------- Kernel 1: per-feature stats -> ws ----------------
// stats[i*8+0]=mean(W[i,:]), +1=mean(b[i,:]), +2=E[Wc^2], +3=E[Wc*bc], +4=E[bc^2]
__device__ __forceinline__ float block_mean128(float v, float* red, int d) {
    red[d] = v;
    __syncthreads();
    for (int s = 64; s > 0; s >>= 1) {
        if (d < s) red[d] += red[d + s];
        __syncthreads();
    }
    float m = red[0] * (1.0f / 128.0f);
    __syncthreads();
    return m;
}

__global__ void te_stats_kernel(const float* __restrict__ W, const float* __restrict__ Bv,
                                float* __restrict__ stats) {
    __shared__ float red[128];
    int i = blockIdx.x;   // feature
    int d = threadIdx.x;  // 0..127
    float w = W[i * ND + d];
    float b = Bv[i * ND + d];
    float mW = block_mean128(w, red, d);
    float mb = block_mean128(b, red, d);
    float wc = w - mW;
    float bc = b - mb;
    float sWW = block_mean128(wc * wc, red, d);
    float sWb = block_mean128(wc * bc, red, d);
    float sbb = block_mean128(bc * bc, red, d);
    if (d == 0) {
        stats[i * 8 + 0] = mW;
        stats[i * 8 + 1] = mb;
        stats[i * 8 + 2] = sWW;
        stats[i * 8 + 3] = sWb;
        stats[i * 8 + 4] = sbb;
    }
}

// ---------------- Kernel 2: streaming WMMA kernel ----------------
// grid = (batch/128, NF/8), block = 256 threads (8 wave32).
// Each wave handles 16 batch rows; block covers 128 rows x 8 features.
#define FPB 8  // features per block

__global__ __launch_bounds__(256) void te_main_kernel(
    const float* __restrict__ x, const float* __restrict__ W,
    const float* __restrict__ Bv, const float* __restrict__ gamma,
    const float* __restrict__ beta, const float* __restrict__ stats,
    float* __restrict__ out) {
    // Interleaved (Wg, bg) pairs so one ds_load_b64 fetches both B VGPRs.
    __shared__ float2 sWB[FPB * ND];  // 8KB

    const int tid = threadIdx.x;
    const int i0 = blockIdx.y * FPB;

    // Stage centered, gamma-scaled rows into LDS.
    for (int k = tid; k < FPB * ND; k += 256) {
        int fi = k >> 7;        // local feature
        int d  = k & 127;
        int gi = i0 + fi;       // global feature
        float g  = gamma[d];
        float mW = stats[gi * 8 + 0];
        float mb = stats[gi * 8 + 1];
        float2 wb;
        wb.x = (W[gi * ND + d] - mW) * g;
        wb.y = (Bv[gi * ND + d] - mb) * g;
        sWB[k] = wb;
    }
    __syncthreads();

    const int lane = tid & 31;
    const int wave = tid >> 5;
    const int ln   = lane & 15;
    const bool lo  = lane < 16;

    const int rowBase = blockIdx.x * 128 + wave * 16;  // 16 rows per wave
    const int rowL    = rowBase + ln;                  // this lane's A-row

    // beta per d-tile (C matrix: value depends only on N = lane&15)
    float cbeta[8];
#pragma unroll
    for (int t = 0; t < 8; t++) cbeta[t] = beta[t * 16 + ln];

    for (int fi = 0; fi < FPB; fi++) {
        const int gi = i0 + fi;
        const float sWW = stats[gi * 8 + 2];
        const float sWb = stats[gi * 8 + 3];
        const float sbb = stats[gi * 8 + 4];

        const float xv  = x[rowL * NF + gi];
        const float var = fmaf(xv, fmaf(xv, sWW, 2.0f * sWb), sbb);
        const float rstd = rsqrtf(var + LN_EPS);
        const float a = xv * rstd;

        // A (16x4 f32): VGPR0 lanes0-15 = K0 (a_m), VGPR1 lanes0-15 = K1 (rstd_m),
        // lanes16-31 (K2,K3) = 0 per ISA 16x4 layout.
        v2f A;
        A.x = lo ? a : 0.0f;
        A.y = lo ? rstd : 0.0f;

        const float2* wbi = sWB + fi * ND;
        float* op = out + (size_t)(rowBase + (lo ? 0 : 8)) * (NF * ND) + (size_t)gi * ND + ln;

#pragma unroll
        for (int t = 0; t < 8; t++) {
            // Unconditional ds_load_b64 (address valid for all lanes: lane&15),
            // then cndmask the value: avoids EXEC save/restore around the load.
            float2 wb = wbi[t * 16 + ln];
            // B (4x16 f32): rows striped across lanes per VGPR; K=0 row (Wg) in
            // VGPR0 lanes0-15, K=1 row (bg) in VGPR1 lanes0-15; K=2,3 rows zero.
            v2f Bm;
            Bm.x = lo ? wb.x : 0.0f;
            Bm.y = lo ? wb.y : 0.0f;

            v8f c;
#pragma unroll
            for (int r = 0; r < 8; r++) c[r] = cbeta[t];

            // D = A(16x4) x B(4x16) + C : one normalized 16x16 output tile
            v8f dacc = __builtin_amdgcn_wmma_f32_16x16x4_f32(
                false, A, false, Bm, (short)0, c, false, false);

            float* opt = op + t * 16;
#pragma unroll
            for (int r = 0; r < 8; r++) {
                // row = rowBase + (lo?0:8) + r ; col = t*16 + ln. NT: stream past L2.
                __builtin_nontemporal_store(dacc[r], opt + (size_t)r * (NF * ND));
            }
        }
    }
}

extern "C" void kernel_launch(void* const* d_in, const int* in_sizes, int n_in,
                              void* d_out, int out_size, void* d_ws, size_t ws_size,
                              hipStream_t stream) {
    const float* x     = (const float*)d_in[0];  // [32768, 64]
    const float* W     = (const float*)d_in[1];  // [64, 128]
    const float* Bv    = (const float*)d_in[2];  // [64, 128]
    const float* gamma = (const float*)d_in[3];  // [128]
    const float* beta  = (const float*)d_in[4];  // [128]
    float* out   = (float*)d_out;                // [32768, 64, 128]
    float* stats = (float*)d_ws;                 // 64*8 floats = 2KB

    const int batch = in_sizes[0] / NF;          // 32768

    te_stats_kernel<<<NF, 128, 0, stream>>>(W, Bv, stats);
    dim3 grid(batch / 128, NF / FPB);            // (256, 8)
    te_main_kernel<<<grid, 256, 0, stream>>>(x, W, Bv, gamma, beta, stats, out);
}